// MultiBoxLoss_84765474554203
// MI455X (gfx1250) — compile-verified
//
#include <hip/hip_runtime.h>
#include <hip/hip_bf16.h>
#include <stdint.h>

#define NP     8732
#define NCLS   21
#define NOBJ   16
#define TPB    512
#define NWAVE  (TPB / 32)
#define NCHUNK ((NP + TPB - 1) / TPB)   // 18

// ---- CDNA5 async copies: global -> LDS, tracked by ASYNCcnt ----------------
__device__ __forceinline__ void async_dbox_to_lds(const float* gbase, int p, void* ldsdst) {
  unsigned lds_addr = (unsigned)(uintptr_t)ldsdst;   // low 32 bits of generic LDS ptr = LDS byte offset
  unsigned voff     = (unsigned)(p << 4);            // prior row = 16B, 16B aligned
  asm volatile("global_load_async_to_lds_b128 %0, %1, %2"
               :: "v"(lds_addr), "v"(voff), "s"(gbase) : "memory");
}

// 21 dword async loads of one conf row; INST_OFFSET shifts BOTH the LDS and the
// global address (CDNA5 ISA), so one (lds,voff) pair serves all 21.
template <int K>
__device__ __forceinline__ void conf_async_rec(unsigned lds0, unsigned v0, const float* gbase) {
  if constexpr (K < NCLS) {
    asm volatile("global_load_async_to_lds_b32 %0, %1, %2 offset:%3"
                 :: "v"(lds0), "v"(v0), "s"(gbase), "n"(K * 4) : "memory");
    conf_async_rec<K + 1>(lds0, v0, gbase);
  }
}

__device__ __forceinline__ void wait_async_le22() { asm volatile("s_wait_asynccnt 22" ::: "memory"); }
__device__ __forceinline__ void wait_async_le1()  { asm volatile("s_wait_asynccnt 1"  ::: "memory"); }
__device__ __forceinline__ void wait_async_le0()  { asm volatile("s_wait_asynccnt 0"  ::: "memory"); }

__device__ __forceinline__ float block_sum_f(float v, float* sw, int tid) {
  int lane = tid & 31, w = tid >> 5;
#pragma unroll
  for (int o = 16; o > 0; o >>= 1) v += __shfl_down(v, o, 32);
  if (lane == 0) sw[w] = v;
  __syncthreads();
  if (w == 0) {
    float x = (lane < NWAVE) ? sw[lane] : 0.f;
#pragma unroll
    for (int o = 8; o > 0; o >>= 1) x += __shfl_down(x, o, 32);
    if (lane == 0) sw[NWAVE] = x;
  }
  __syncthreads();
  float r = sw[NWAVE];
  __syncthreads();
  return r;
}

__device__ __forceinline__ int block_sum_i(int v, int* sw, int tid) {
  int lane = tid & 31, w = tid >> 5;
#pragma unroll
  for (int o = 16; o > 0; o >>= 1) v += __shfl_down(v, o, 32);
  if (lane == 0) sw[w] = v;
  __syncthreads();
  if (w == 0) {
    int x = (lane < NWAVE) ? sw[lane] : 0;
#pragma unroll
    for (int o = 8; o > 0; o >>= 1) x += __shfl_down(x, o, 32);
    if (lane == 0) sw[NWAVE] = x;
  }
  __syncthreads();
  int r = sw[NWAVE];
  __syncthreads();
  return r;
}

__device__ __forceinline__ float smooth_l1(float d) {
  float ad = fabsf(d);
  return (ad < 1.f) ? 0.5f * d * d : ad - 0.5f;
}

__global__ __launch_bounds__(TPB)
void mbox_loss_kernel(const float* __restrict__ loc_data,
                      const float* __restrict__ conf_data,
                      const float* __restrict__ dbox,
                      const float* __restrict__ targets,
                      float* __restrict__ part) {
  // ~143 KB LDS (CDNA5: up to 320 KB per workgroup). s_btv is reused as the
  // HNM loss array in pass 2 (safe: prior p read then overwritten by one thread).
  __shared__ float          s_btv[NP];        // best_truth_overlap -> later loss_c_hnm
  __shared__ unsigned char  s_bti[NP];        // best_truth_idx (0..15)
  __shared__ float          s_tru[NOBJ * 4];  // truths, corner form
  __shared__ float          s_lab[NOBJ];
  __shared__ int            s_bpi[NOBJ];      // best_prior_idx per truth
  __shared__ float          s_swf[NWAVE + 1];
  __shared__ int            s_swi[NWAVE + 1];
  alignas(16) __shared__ float s_dbox[2][TPB * 4];     // async-staged priors
  alignas(16) __shared__ float s_conf[2][TPB * NCLS];  // async-staged conf rows (2x43KB)

  const int b   = blockIdx.x;
  const int tid = threadIdx.x;
  const int lane = tid & 31, wv = tid >> 5;
  const float* cbase = conf_data + (size_t)b * NP * NCLS;

  // ---- load truths/labels ----
  if (tid < NOBJ * 5) {
    int o = tid / 5, k = tid % 5;
    float v = targets[((size_t)b * NOBJ + o) * 5 + k];
    if (k < 4) s_tru[o * 4 + k] = v; else s_lab[o] = v;
  }
  __syncthreads();

  // ---- pass 1: matching (IoU), async-pipelined dbox staging ----
  float tv[NOBJ]; int ti[NOBJ];
#pragma unroll
  for (int t = 0; t < NOBJ; ++t) { tv[t] = -1.f; ti[t] = 0x7fffffff; }

  {
    int p0 = tid; if (p0 > NP - 1) p0 = NP - 1;
    async_dbox_to_lds(dbox, p0, &s_dbox[0][tid * 4]);
  }
  for (int c = 0; c < NCHUNK; ++c) {
    if (c + 1 < NCHUNK) {
      int pn = (c + 1) * TPB + tid; if (pn > NP - 1) pn = NP - 1;
      async_dbox_to_lds(dbox, pn, &s_dbox[(c + 1) & 1][tid * 4]);
      wait_async_le1();
    } else {
      wait_async_le0();
    }
    int p = c * TPB + tid;
    if (p < NP) {
      const float* dp = &s_dbox[c & 1][tid * 4];
      float cx = dp[0], cy = dp[1], w = dp[2], h = dp[3];
      float px0 = cx - w * 0.5f, py0 = cy - h * 0.5f;
      float px1 = cx + w * 0.5f, py1 = cy + h * 0.5f;
      float area_b = (px1 - px0) * (py1 - py0);
      float bv = -1.f; int bt = 0;
#pragma unroll
      for (int t = 0; t < NOBJ; ++t) {
        float ax0 = s_tru[t * 4 + 0], ay0 = s_tru[t * 4 + 1];
        float ax1 = s_tru[t * 4 + 2], ay1 = s_tru[t * 4 + 3];
        float lx = fmaxf(ax0, px0), ly = fmaxf(ay0, py0);
        float rx = fminf(ax1, px1), ry = fminf(ay1, py1);
        float iw = fmaxf(rx - lx, 0.f), ih = fmaxf(ry - ly, 0.f);
        float inter = iw * ih;
        float aa = (ax1 - ax0) * (ay1 - ay0);
        float iou = inter / (aa + area_b - inter);
        if (iou > bv) { bv = iou; bt = t; }             // first-occurrence argmax over truths
        if (iou > tv[t]) { tv[t] = iou; ti[t] = p; }    // ascending p => first occurrence kept
      }
      s_btv[p] = bv;
      s_bti[p] = (unsigned char)bt;
    }
  }
  __syncthreads();

  // ---- per-truth argmax over priors (ties -> smaller prior idx) ----
  for (int t = 0; t < NOBJ; ++t) {
    float v = tv[t]; int i = ti[t];
#pragma unroll
    for (int o = 16; o > 0; o >>= 1) {
      float ov = __shfl_down(v, o, 32); int oi = __shfl_down(i, o, 32);
      if (ov > v || (ov == v && oi < i)) { v = ov; i = oi; }
    }
    if (lane == 0) { s_swf[wv] = v; s_swi[wv] = i; }
    __syncthreads();
    if (wv == 0) {
      float x = (lane < NWAVE) ? s_swf[lane] : -2.f;
      int  xi = (lane < NWAVE) ? s_swi[lane] : 0x7fffffff;
#pragma unroll
      for (int o = 8; o > 0; o >>= 1) {
        float ov = __shfl_down(x, o, 32); int oi = __shfl_down(xi, o, 32);
        if (ov > x || (ov == x && oi < xi)) { x = ov; xi = oi; }
      }
      if (lane == 0) s_bpi[t] = xi;
    }
    __syncthreads();
  }

  // ---- force-match override (sequential, last write wins like scatter) ----
  if (tid == 0) {
    for (int t = 0; t < NOBJ; ++t) {
      int p = s_bpi[t];
      s_btv[p] = 2.0f;
      s_bti[p] = (unsigned char)t;
    }
  }
  __syncthreads();

  // ---- pass 2: encode + smooth-L1 + CE; conf rows async-staged to LDS ----
  float lossl = 0.f, cepos = 0.f; int npos = 0;
  {
    int p0 = tid; if (p0 > NP - 1) p0 = NP - 1;
    async_dbox_to_lds(dbox, p0, &s_dbox[0][tid * 4]);
    conf_async_rec<0>((unsigned)(uintptr_t)&s_conf[0][tid * NCLS],
                      (unsigned)(p0 * (NCLS * 4)), cbase);
  }
  for (int c = 0; c < NCHUNK; ++c) {
    if (c + 1 < NCHUNK) {
      int pn = (c + 1) * TPB + tid; if (pn > NP - 1) pn = NP - 1;
      async_dbox_to_lds(dbox, pn, &s_dbox[(c + 1) & 1][tid * 4]);
      conf_async_rec<0>((unsigned)(uintptr_t)&s_conf[(c + 1) & 1][tid * NCLS],
                        (unsigned)(pn * (NCLS * 4)), cbase);
      wait_async_le22();   // 22 newest ops (next chunk) may remain outstanding
    } else {
      wait_async_le0();
    }
    int p = c * TPB + tid;
    if (p < NP) {
      int   t   = (int)s_bti[p];
      float btv = s_btv[p];
      int confi = (btv < 0.5f) ? 0 : ((int)s_lab[t] + 1);

      const float* cr = &s_conf[c & 1][tid * NCLS];
      float x[NCLS];
#pragma unroll
      for (int k = 0; k < NCLS; ++k) x[k] = cr[k];
      float m = x[0];
#pragma unroll
      for (int k = 1; k < NCLS; ++k) m = fmaxf(m, x[k]);
      float ssum = 0.f, xs = 0.f;
#pragma unroll
      for (int k = 0; k < NCLS; ++k) {
        ssum += expf(x[k] - m);
        xs = (k == confi) ? x[k] : xs;
      }
      float ce = (m + logf(ssum)) - xs;

      if (confi > 0) {
        npos += 1; cepos += ce;
        const float* dp = &s_dbox[c & 1][tid * 4];
        float cx = dp[0], cy = dp[1], pw = dp[2], ph = dp[3];
        float m0 = s_tru[t * 4 + 0], m1 = s_tru[t * 4 + 1];
        float m2 = s_tru[t * 4 + 2], m3 = s_tru[t * 4 + 3];
        float g0 = ((m0 + m2) * 0.5f - cx) / (0.1f * pw);
        float g1 = ((m1 + m3) * 0.5f - cy) / (0.1f * ph);
        float g2 = logf((m2 - m0) / pw) / 0.2f;
        float g3 = logf((m3 - m1) / ph) / 0.2f;
        const float4 ld = *reinterpret_cast<const float4*>(loc_data + ((size_t)b * NP + p) * 4);
        lossl += smooth_l1(ld.x - g0) + smooth_l1(ld.y - g1)
               + smooth_l1(ld.z - g2) + smooth_l1(ld.w - g3);
      }
      s_btv[p] = (confi > 0) ? 0.f : ce;   // loss_c_hnm (>= 0)
    }
  }
  __syncthreads();

  lossl = block_sum_f(lossl, s_swf, tid);
  cepos = block_sum_f(cepos, s_swf, tid);
  npos  = block_sum_i(npos,  s_swi, tid);

  int k = npos * 3; if (k > NP) k = NP;

  // ---- exact top-k sum via radix-select on float bits (all values >= 0) ----
  float lossc = cepos;
  if (k > 0) {
    unsigned prefix = 0u; int r = k;
    for (int bit = 30; bit >= 0; --bit) {
      unsigned maskhi = ~((1u << bit) - 1u);
      unsigned want   = prefix | (1u << bit);
      int cnt = 0;
      for (int c = 0; c < NCHUNK; ++c) {
        int p = c * TPB + tid;
        if (p < NP) {
          unsigned vb = __float_as_uint(s_btv[p]);
          cnt += ((vb & maskhi) == want) ? 1 : 0;
        }
      }
      cnt = block_sum_i(cnt, s_swi, tid);
      if (cnt >= r) prefix = want; else r -= cnt;
    }
    float tau = __uint_as_float(prefix);   // k-th largest value
    float sgt = 0.f; int cgt = 0;
    for (int c = 0; c < NCHUNK; ++c) {
      int p = c * TPB + tid;
      if (p < NP) {
        float v = s_btv[p];
        if (__float_as_uint(v) > prefix) { sgt += v; cgt += 1; }
      }
    }
    sgt = block_sum_f(sgt, s_swf, tid);
    cgt = block_sum_i(cgt, s_swi, tid);
    lossc = cepos + sgt + (float)(k - cgt) * tau;  // ties at tau contribute identically
  }

  if (tid == 0) {
    part[b * 3 + 0] = lossl;
    part[b * 3 + 1] = lossc;
    part[b * 3 + 2] = (float)npos;
  }
}

__global__ void mbox_finalize(const float* __restrict__ part, int B, float* __restrict__ out) {
  if (blockIdx.x == 0 && threadIdx.x == 0) {
    float L = 0.f, C = 0.f, N = 0.f;
    for (int i = 0; i < B; ++i) {          // fixed order -> deterministic
      L += part[i * 3 + 0];
      C += part[i * 3 + 1];
      N += part[i * 3 + 2];
    }
    out[0] = L / N;
    out[1] = C / N;
  }
}

extern "C" void kernel_launch(void* const* d_in, const int* in_sizes, int n_in,
                              void* d_out, int out_size, void* d_ws, size_t ws_size,
                              hipStream_t stream) {
  const float* loc  = (const float*)d_in[0];
  const float* conf = (const float*)d_in[1];
  const float* dbx  = (const float*)d_in[2];
  const float* targ = (const float*)d_in[3];
  int Np = in_sizes[2] / 4;                 // 8732 (kernel specialized for this)
  int B  = in_sizes[0] / (Np * 4);          // 128
  float* part = (float*)d_ws;               // B*3 floats of scratch

  mbox_loss_kernel<<<B, TPB, 0, stream>>>(loc, conf, dbx, targ, part);
  mbox_finalize<<<1, 32, 0, stream>>>(part, B, (float*)d_out);
}